// RealToComplex_24584392802924
// MI455X (gfx1250) — compile-verified
//
#include <hip/hip_runtime.h>

// ---------------------------------------------------------------------------
// Phase-corrected real/imag reconstruction, MI455X (gfx1250).
// x: (16, 128, 16384) f32  ->  out: (16, 64, 16384) complex64 (interleaved)
// Bandwidth-bound (~268MB @ 23.3TB/s ~= 11.5us). Strategy:
//   * TDM (tensor_load_to_lds) DMAs each phase row into LDS, with TDM pad
//     insertion (4 dwords per 64) so per-thread chunked LDS reads are ~conflict-free.
//   * Double-buffered rows, prefetch row i+2 while computing row i
//     (s_wait_tensorcnt pipelining). 32 waves each DMA their own 512-f32 slice.
//   * Cumsum of +-1 jump corrections: per-thread serial chunk (16) + block
//     Hillis-Steele scan in LDS (wave32, 1024 threads).
// ---------------------------------------------------------------------------

#define N_ROW    16384
#define HALF_CH  64
#define N_BATCH  16
#define N_ROWS_T (N_BATCH * HALF_CH)               // 1024 rows total
#define BLOCK    1024
#define CHUNK    (N_ROW / BLOCK)                   // 16 elements per thread
#define ROWS_WG  4
#define NUM_WG   (N_ROWS_T / ROWS_WG)              // 256 workgroups
#define SLICE    (N_ROW / 32)                      // 512 floats per wave slice
#define PAD_SEG  64                                // pad after every 64 floats
#define PAD_AMT  4                                 // 4 floats of pad (16B, keeps b128 align)
#define ROW_LDSF (N_ROW + (N_ROW / PAD_SEG) * PAD_AMT)     // 17408 floats
#define SLICE_LDSF (SLICE + (SLICE / PAD_SEG) * PAD_AMT)   // 544 floats

typedef unsigned int u32;
typedef u32 __attribute__((ext_vector_type(4))) u32x4_t;
typedef int  __attribute__((ext_vector_type(4))) i32x4_t;
typedef int  __attribute__((ext_vector_type(8))) i32x8_t;

#if __has_builtin(__builtin_amdgcn_tensor_load_to_lds) && __has_builtin(__builtin_amdgcn_s_wait_tensorcnt)
#define USE_TDM 1
#else
#define USE_TDM 0
#endif

// padded LDS float index: insert 4 floats of pad every 64 floats
__device__ __forceinline__ int pf(int g) { return g + ((g >> 6) << 2); }

// d = sign(grad) where |grad| > 0.5, else 0
__device__ __forceinline__ float dval(float g) {
    return (g > 0.5f) ? 1.0f : ((g < -0.5f) ? -1.0f : 0.0f);
}

__device__ __forceinline__ const float* phase_row_ptr(const float* x, int row) {
    int b = row >> 6, c = row & (HALF_CH - 1);
    return x + ((size_t)b * (2 * HALF_CH) + HALF_CH + c) * (size_t)N_ROW;
}

#if USE_TDM
// 1D TDM copy: SLICE f32 from global -> LDS, padding 4 dwords per 64 dwords.
// D# group0: count=1 | lds_addr | global_addr | type=2
// D# group1: data_size=4B, pad_enable, pad_interval=64dw(code 5), pad_amount=4dw(code 3),
//            tensor_dim0=tile_dim0=SLICE, tensor_dim0_stride=SLICE
__device__ __forceinline__ void tdm_load_slice(const float* gsrc, u32 lds_byte)
{
    unsigned long long ga = (unsigned long long)(uintptr_t)gsrc;
    u32x4_t g0;
    g0.x = 1u;                                         // count = 1 valid descriptor
    g0.y = lds_byte;                                   // lds_addr [63:32]
    g0.z = (u32)ga;                                    // global_addr[31:0]
    g0.w = (u32)((ga >> 32) & 0x1FFFFFFull) | (2u << 30); // global_addr[56:32] | type=2

    i32x8_t g1;
    g1[0] = (int)((2u << 16) | (1u << 20) | (5u << 22) | (3u << 25));
    g1[1] = (int)(((u32)SLICE & 0xFFFFu) << 16);       // tensor_dim0[15:0] @ [63:48]
    g1[2] = (int)(1u << 16);                           // tensor_dim0[31:16]=0, tensor_dim1=1
    g1[3] = (int)(((u32)SLICE & 0xFFFFu) << 16);       // tile_dim0 @ [127:112]
    g1[4] = 0;                                         // tile_dim1=0, tile_dim2=0
    g1[5] = (int)SLICE;                                // tensor_dim0_stride[31:0]
    g1[6] = 0;
    g1[7] = 0;

    i32x4_t z4 = {0, 0, 0, 0};
#if defined(__clang_major__) && (__clang_major__ >= 23)
    i32x8_t z8 = {0, 0, 0, 0, 0, 0, 0, 0};
    __builtin_amdgcn_tensor_load_to_lds(g0, g1, z4, z4, z8, 0);
#else
    __builtin_amdgcn_tensor_load_to_lds(g0, g1, z4, z4, 0);
#endif
}
#endif

__global__ __launch_bounds__(BLOCK)
void phase_correct_kernel(const float* __restrict__ x, float* __restrict__ out)
{
    __shared__ float lph[2][ROW_LDSF];   // double-buffered padded phase rows (136KB)
    __shared__ float lscan[BLOCK];       // block scan workspace (4KB)

    const int t    = threadIdx.x;
    const int wv   = t >> 5;             // wave id (wave32)
    const int row0 = blockIdx.x * ROWS_WG;

#if USE_TDM
    {   // prologue: every wave DMAs its slice of rows 0 and 1 (uniform per-wave)
        u32 base0 = (u32)(uintptr_t)(&lph[0][0]) + (u32)(wv * SLICE_LDSF * 4);
        u32 base1 = (u32)(uintptr_t)(&lph[1][0]) + (u32)(wv * SLICE_LDSF * 4);
        tdm_load_slice(phase_row_ptr(x, row0 + 0) + wv * SLICE, base0);
        tdm_load_slice(phase_row_ptr(x, row0 + 1) + wv * SLICE, base1);
    }
#endif

    for (int i = 0; i < ROWS_WG; ++i) {
        const int row = row0 + i;
        const int b = row >> 6, c = row & (HALF_CH - 1);
        const float* mrow = x + ((size_t)b * (2 * HALF_CH) + c) * (size_t)N_ROW;
        float* orow = out + (size_t)row * (2 * (size_t)N_ROW);
        float* buf = &lph[i & 1][0];

#if USE_TDM
        if (i == ROWS_WG - 1) __builtin_amdgcn_s_wait_tensorcnt(0);
        else                  __builtin_amdgcn_s_wait_tensorcnt(1);
        __syncthreads();
#else
        {   // fallback: cooperative coalesced load with software padding
            const float4* prow4 = (const float4*)phase_row_ptr(x, row);
            #pragma unroll
            for (int it = 0; it < (N_ROW / 4) / BLOCK; ++it) {
                int q = t + BLOCK * it;
                float4 v = prow4[q];
                int p = pf(q * 4);
                buf[p] = v.x; buf[p + 1] = v.y; buf[p + 2] = v.z; buf[p + 3] = v.w;
            }
        }
        __syncthreads();
#endif
        const int a = t * CHUNK;

        // load this thread's 16 phase values from padded LDS (b128 reads)
        float pn[CHUNK];
        {
            const float4* bp = (const float4*)&buf[pf(a)];
            #pragma unroll
            for (int q = 0; q < CHUNK / 4; ++q) {
                float4 v = bp[q];
                pn[4*q+0] = v.x; pn[4*q+1] = v.y; pn[4*q+2] = v.z; pn[4*q+3] = v.w;
            }
        }
        float pn_m1 = (a > 0) ? buf[pf(a - 1)] : pn[0];   // a==0 -> grad 0 -> d 0
        float pn_m2 = (a > 1) ? buf[pf(a - 2)] : 0.0f;
        float pn_nx = (a + CHUNK < N_ROW) ? buf[pf(a + CHUNK)] : 0.0f;

        // pass 1: local sum of +-1 jump corrections
        float lsum = 0.0f;
        {
            float prev = pn_m1;
            #pragma unroll
            for (int k = 0; k < CHUNK; ++k) {
                lsum += dval(pn[k] - prev);
                prev = pn[k];
            }
        }

        // block-wide inclusive scan (Hillis-Steele, 10 rounds)
        lscan[t] = lsum;
        __syncthreads();
        float val = lsum;
        #pragma unroll
        for (int off = 1; off < BLOCK; off <<= 1) {
            float other = (t >= off) ? lscan[t - off] : 0.0f;
            __syncthreads();
            val += other;
            lscan[t] = val;
            __syncthreads();
        }
        const float excl = val - lsum;   // S[a-1] = sum of d over [0, a)

        // magnitudes (per-thread chunked float4 global loads; all lines consumed)
        float mg[CHUNK];
        {
            const float4* mp = (const float4*)(mrow + a);
            #pragma unroll
            for (int q = 0; q < CHUNK / 4; ++q) {
                float4 v = mp[q];
                mg[4*q+0] = v.x; mg[4*q+1] = v.y; mg[4*q+2] = v.z; mg[4*q+3] = v.w;
            }
        }

        const float TWO_PI_F = 6.283185307179586f;
        const float PI_F     = 3.1415926535897932f;

        // phase[a-1] for the smoothing halo
        float ph_prev;
        if (a == 0) {
            ph_prev = 0.0f;                               // zero padding at row start
        } else {
            float d_am1 = dval(pn_m1 - pn_m2);            // a-1 >= 15 here, never j==0
            ph_prev = (pn_m1 - (excl - d_am1)) * TWO_PI_F - PI_F;
        }
        float S_run  = excl;                              // S[j-1] at j = a
        float ph_cur = (pn[0] - S_run) * TWO_PI_F - PI_F;

        #pragma unroll
        for (int k = 0; k < CHUNK; ++k) {
            const int j = a + k;
            float pprev = (k == 0) ? pn_m1 : pn[k - 1];
            float dj    = (j == 0) ? 0.0f : dval(pn[k] - pprev);
            float S_j   = S_run + dj;                     // S[j]
            float pnn   = (k == CHUNK - 1) ? pn_nx : pn[k + 1];
            float ph_next = (j + 1 < N_ROW) ? ((pnn - S_j) * TWO_PI_F - PI_F) : 0.0f;

            float sm  = (ph_prev + ph_cur + ph_next) * (1.0f / 3.0f);
            float phf = 0.7f * ph_cur + 0.3f * sm;
            float s, co;
            __sincosf(phf, &s, &co);
            float2 o;
            o.x = mg[k] * co;
            o.y = mg[k] * s;
            ((float2*)orow)[j] = o;                       // interleaved complex64

            ph_prev = ph_cur; ph_cur = ph_next; S_run = S_j;
        }

        __syncthreads();   // all reads of buf done before DMA reuses it

#if USE_TDM
        if (i + 2 < ROWS_WG) {   // prefetch row i+2 into the buffer just freed
            u32 basen = (u32)(uintptr_t)(&lph[i & 1][0]) + (u32)(wv * SLICE_LDSF * 4);
            tdm_load_slice(phase_row_ptr(x, row0 + i + 2) + wv * SLICE, basen);
        }
#endif
    }
}

extern "C" void kernel_launch(void* const* d_in, const int* in_sizes, int n_in,
                              void* d_out, int out_size, void* d_ws, size_t ws_size,
                              hipStream_t stream) {
    (void)in_sizes; (void)n_in; (void)out_size; (void)d_ws; (void)ws_size;
    const float* x = (const float*)d_in[0];
    float* out = (float*)d_out;
    phase_correct_kernel<<<NUM_WG, BLOCK, 0, stream>>>(x, out);
}